// MemoryGraph_23227183137302
// MI455X (gfx1250) — compile-verified
//
#include <hip/hip_runtime.h>
#include <hip/hip_bf16.h>
#include <math.h>

#define NN 4096
#define DD 64
#define BB 8
#define CC 64

typedef __attribute__((ext_vector_type(2))) float v2f;
typedef __attribute__((ext_vector_type(8))) float v8f;

// ---- CDNA5 fast-math paths (guarded; fall back to libm if builtin absent) ----
__device__ __forceinline__ float fast_tanh(float x) {
#if __has_builtin(__builtin_amdgcn_tanhf)
    return __builtin_amdgcn_tanhf(x);          // v_tanh_f32
#elif __has_builtin(__builtin_amdgcn_tanh_f32)
    return __builtin_amdgcn_tanh_f32(x);
#else
    return tanhf(x);
#endif
}

__device__ __forceinline__ float fast_sigmoid(float x) {
#if __has_builtin(__builtin_amdgcn_exp2f)
    return 1.0f / (1.0f + __builtin_amdgcn_exp2f(-1.4426950408889634f * x)); // v_exp_f32
#else
    return 1.0f / (1.0f + expf(-x));
#endif
}

// ---- async global->LDS copy (CDNA5 GLOBAL_LOAD_ASYNC_TO_LDS_B128), guarded ----
#if defined(__has_builtin)
#if __has_builtin(__builtin_amdgcn_global_load_async_to_lds_b128) && \
    __has_builtin(__builtin_amdgcn_s_wait_asynccnt)
#define USE_ASYNC_LDS 1
#endif
#endif
#ifndef USE_ASYNC_LDS
#define USE_ASYNC_LDS 0
#endif

typedef int v4i __attribute__((vector_size(4 * sizeof(int))));
typedef __attribute__((address_space(1))) v4i as1_v4i;   // global
typedef __attribute__((address_space(3))) v4i as3_v4i;   // LDS

__device__ __forceinline__ void async_cp16(const float* g, float* l) {
#if USE_ASYNC_LDS
    __builtin_amdgcn_global_load_async_to_lds_b128(
        (as1_v4i*)(unsigned long long)g,
        (as3_v4i*)(unsigned)(unsigned long long)l,
        0, 0);
#endif
}

__device__ __forceinline__ void async_wait() {
#if USE_ASYNC_LDS
    __builtin_amdgcn_s_wait_asynccnt(0);
#endif
}

// One wave computes Y(16x64) = act(X(16xK) * W(Kx64) + bias), X/Y in LDS, W/bias global.
// WMMA f32 16x16x4: A 16x4 (lanes0-15: K=k0,k0+1 ; lanes16-31: K=k0+2,k0+3),
// C/D 16x16 (VGPR r: lanes0-15 -> M=r, lanes16-31 -> M=r+8, N=lane&15).
template <int K, bool ACT_TANH>
__device__ __forceinline__ void wave_gemm64(const float* __restrict__ X, int ldx,
                                            const float* __restrict__ W,
                                            const float* __restrict__ bias,
                                            float* __restrict__ Y, int ldy,
                                            int lane)
{
    const int l16  = lane & 15;
    const int sel  = lane >> 4;        // 0: lanes 0-15, 1: lanes 16-31
    const int koff = sel * 2;
#pragma unroll
    for (int ct = 0; ct < 4; ++ct) {
        const int col = ct * 16 + l16;
        v8f c = {};
#pragma unroll
        for (int k0 = 0; k0 < K; k0 += 4) {
            v2f a, b;
            a.x = X[l16 * ldx + k0 + koff];
            a.y = X[l16 * ldx + k0 + koff + 1];
            b.x = W[(size_t)(k0 + koff) * 64 + col];
            b.y = W[(size_t)(k0 + koff + 1) * 64 + col];
            c = __builtin_amdgcn_wmma_f32_16x16x4_f32(false, a, false, b,
                                                      (short)0, c, false, false);
        }
        const float bb = bias[col];
#pragma unroll
        for (int r = 0; r < 8; ++r) {
            float v = c[r] + bb;
            if (ACT_TANH) v = fast_tanh(v);
            Y[(r + sel * 8) * ldy + col] = v;
        }
    }
}

// ---------------- Phase 1: prev_msg = tanh(msgMLP(h)) for all N nodes ----------------
__global__ void __launch_bounds__(128)
k_prev_msg(const float* __restrict__ h,
           const float* __restrict__ w1, const float* __restrict__ b1,
           const float* __restrict__ w2, const float* __restrict__ b2,
           float* __restrict__ pm)
{
    __shared__ __align__(16) float lds[4 * 2 * 16 * 68];
    const int wave = threadIdx.x >> 5;
    const int lane = threadIdx.x & 31;
    const int n    = blockIdx.x * 4 + wave;
    float* X = lds + wave * (2 * 16 * 68);
    float* Y = X + 16 * 68;

    // warm L2 with the second-layer weights while GEMM-1 streams w1
    for (int i = lane * 32; i < 4096; i += 32 * 32)
        __builtin_prefetch(w2 + (size_t)n * 4096 + i, 0, 1);

    // stage X rows: b=0..7 from h, rows 8..15 zero
#if USE_ASYNC_LDS
    for (int i = lane; i < 8 * 16; i += 32) {
        const int row = i >> 4, q = i & 15;
        async_cp16(h + ((size_t)row * NN + n) * DD + q * 4, X + row * 68 + q * 4);
    }
    for (int i = lane; i < 8 * 16; i += 32) {
        const int row = 8 + (i >> 4), q = i & 15;
        float4 z = {0.f, 0.f, 0.f, 0.f};
        *(float4*)(X + row * 68 + q * 4) = z;
    }
    async_wait();
#else
    for (int i = lane; i < 16 * 16; i += 32) {
        const int row = i >> 4, q = i & 15;
        float4 v = {0.f, 0.f, 0.f, 0.f};
        if (row < 8) v = *(const float4*)(h + ((size_t)row * NN + n) * DD + q * 4);
        *(float4*)(X + row * 68 + q * 4) = v;
    }
#endif
    __syncthreads();
    wave_gemm64<64, true>(X, 68, w1 + (size_t)n * DD * DD, b1 + (size_t)n * DD, Y, 68, lane);
    __syncthreads();
    wave_gemm64<64, true>(Y, 68, w2 + (size_t)n * DD * DD, b2 + (size_t)n * DD, X, 68, lane);
    __syncthreads();
    for (int i = lane; i < 8 * 64; i += 32) {
        const int b = i >> 6, d = i & 63;
        pm[((size_t)b * NN + n) * DD + d] = X[b * 68 + d];
    }
}

// ---------------- Phase 1b: eff_key for nodes C..2C-1 ----------------
__global__ void __launch_bounds__(64)
k_effkey(const float* __restrict__ h, const float* __restrict__ tp, const float* __restrict__ tk,
         const float* __restrict__ w1, const float* __restrict__ b1,
         const float* __restrict__ w2, const float* __restrict__ b2,
         const float* __restrict__ lr_logit, const float* __restrict__ key_p,
         float* __restrict__ ekw)
{
    __shared__ __align__(16) float lds[2 * (16 * 196 + 16 * 68 + 16)];
    const int wave = threadIdx.x >> 5;
    const int lane = threadIdx.x & 31;
    const int node_off = blockIdx.x * 2 + wave;   // 0..63
    const int n = CC + node_off;
    float* X = lds + wave * (16 * 196 + 16 * 68 + 16);
    float* Y = X + 16 * 196;
    float* gateL = Y + 16 * 68;   // 8 (tanh gate)
    float* nrmL  = gateL + 8;     // 8 (reciprocal norm)

    // stage 192-wide input: [h | trace_prim | trace_key]
#if USE_ASYNC_LDS
    for (int i = lane; i < 8 * 48; i += 32) {
        const int row = i / 48, q = i - row * 48;
        const int dd = q * 4;
        const float* src;
        if (dd < 64)       src = h  + ((size_t)row * NN + n) * DD + dd;
        else if (dd < 128) src = tp + ((size_t)row * NN + n) * DD + (dd - 64);
        else               src = tk + ((size_t)row * NN + n) * DD + (dd - 128);
        async_cp16(src, X + row * 196 + q * 4);
    }
    for (int i = lane; i < 8 * 48; i += 32) {
        const int row = 8 + i / 48, q = i % 48;
        float4 z = {0.f, 0.f, 0.f, 0.f};
        *(float4*)(X + row * 196 + q * 4) = z;
    }
    async_wait();
#else
    for (int i = lane; i < 16 * 48; i += 32) {
        const int row = i / 48, q = i - row * 48;
        float4 v = {0.f, 0.f, 0.f, 0.f};
        if (row < 8) {
            const int dd = q * 4;
            const float* src;
            if (dd < 64)       src = h  + ((size_t)row * NN + n) * DD + dd;
            else if (dd < 128) src = tp + ((size_t)row * NN + n) * DD + (dd - 64);
            else               src = tk + ((size_t)row * NN + n) * DD + (dd - 128);
            v = *(const float4*)src;
        }
        *(float4*)(X + row * 196 + q * 4) = v;
    }
#endif
    __syncthreads();
    wave_gemm64<192, true>(X, 196, w1 + (size_t)n * 192 * 64, b1 + (size_t)n * 64, Y, 68, lane);
    __syncthreads();
    if (lane < 8) {
        // second layer: only output column 1 is needed (gate_key)
        float acc = b2[(size_t)n * 3 + 1];
        for (int hh = 0; hh < 64; ++hh)
            acc += Y[lane * 68 + hh] * w2[((size_t)n * 64 + hh) * 3 + 1];
        gateL[lane] = fast_tanh(acc);
        float s = 0.f;
        for (int d = 0; d < 64; ++d) {
            const float t = tk[((size_t)lane * NN + n) * DD + d];
            s += t * t;
        }
        nrmL[lane] = 1.0f / fmaxf(sqrtf(s), 1e-8f);
    }
    __syncthreads();
    const float lr = fast_sigmoid(lr_logit[0]);
    for (int i = lane; i < 512; i += 32) {
        const int b = i >> 6, d = i & 63;
        ekw[(size_t)node_off * 512 + i] =
            key_p[(size_t)n * 64 + d] +
            lr * gateL[b] * tk[((size_t)b * NN + n) * DD + d] * nrmL[b];
    }
}

// ---------------- Phase 2: gather + tree + int-MLP + gated update + msg-MLP ----------------
__global__ void __launch_bounds__(32)
k_update(const float* __restrict__ h, const int* __restrict__ conn,
         const unsigned char* __restrict__ cmask,
         const float* __restrict__ dbw, const float* __restrict__ dgw,
         const float* __restrict__ iw1, const float* __restrict__ ib1,
         const float* __restrict__ iw2, const float* __restrict__ ib2,
         const float* __restrict__ igate,
         const float* __restrict__ mw1, const float* __restrict__ mb1,
         const float* __restrict__ mw2, const float* __restrict__ mb2,
         const float* __restrict__ pm, const float* __restrict__ ekw,
         float* __restrict__ out)
{
    __shared__ __align__(16) float A[16 * 132];
    __shared__ __align__(16) float Bf[16 * 68];
    __shared__ float ek[512];
    __shared__ float cw[32];
    __shared__ int   ci[32];
    const int lane = threadIdx.x;
    const int node_off = blockIdx.x;   // 0..63
    const int n = CC + node_off;

    for (int i = lane; i < 512; i += 32) ek[i] = ekw[(size_t)node_off * 512 + i];
    ci[lane] = conn[(size_t)n * 32 + lane];
    cw[lane] = cmask[(size_t)n * 32 + lane] ? 1.0f : 0.0f;
    // stage h into xin cols 0..63 (rows 8..15 zero)
#if USE_ASYNC_LDS
    for (int i = lane; i < 8 * 16; i += 32) {
        const int row = i >> 4, q = i & 15;
        async_cp16(h + ((size_t)row * NN + n) * DD + q * 4, A + row * 132 + q * 4);
    }
    for (int i = lane; i < 8 * 16; i += 32) {
        const int row = 8 + (i >> 4), q = i & 15;
        float4 z = {0.f, 0.f, 0.f, 0.f};
        *(float4*)(A + row * 132 + q * 4) = z;
    }
    async_wait();
#else
    for (int i = lane; i < 16 * 16; i += 32) {
        const int row = i >> 4, q = i & 15;
        float4 v = {0.f, 0.f, 0.f, 0.f};
        if (row < 8) v = *(const float4*)(h + ((size_t)row * NN + n) * DD + q * 4);
        *(float4*)(A + row * 132 + q * 4) = v;
    }
#endif
    __syncthreads();

    // lane <-> (batch b, d-quarter dq); each lane owns 16 d's
    const int b = lane >> 2, dq = lane & 3;
    const float* ekrow = ek + b * 64 + dq * 16;
    float gacc[16];
#pragma unroll
    for (int t = 0; t < 16; ++t) gacc[t] = 0.f;

    for (int j = 0; j < 4; ++j) {          // NB branches
        float bacc[16];
#pragma unroll
        for (int t = 0; t < 16; ++t) bacc[t] = 0.f;
        for (int i = 0; i < 8; ++i) {      // BRANCH
            const int k   = j * 8 + i;
            const int idx = ci[k];
            const float* gp = pm + ((size_t)b * NN + idx) * DD + dq * 16;
            float4 g0 = *(const float4*)(gp + 0);
            float4 g1 = *(const float4*)(gp + 4);
            float4 g2 = *(const float4*)(gp + 8);
            float4 g3 = *(const float4*)(gp + 12);
            float gv[16] = {g0.x, g0.y, g0.z, g0.w, g1.x, g1.y, g1.z, g1.w,
                            g2.x, g2.y, g2.z, g2.w, g3.x, g3.y, g3.z, g3.w};
            float p = 0.f;
#pragma unroll
            for (int t = 0; t < 16; ++t) p += ekrow[t] * gv[t];
            p += __shfl_xor(p, 1, 32);     // reduce across the 4 lanes of this batch
            p += __shfl_xor(p, 2, 32);
            const float s = fast_tanh(p) * cw[k];
            const float* bw = dbw + (((size_t)n * 4 + j) * 8 + i) * DD + dq * 16;
#pragma unroll
            for (int t = 0; t < 16; ++t) bacc[t] += gv[t] * s * bw[t];
        }
        const float* gw = dgw + ((size_t)n * 4 + j) * DD + dq * 16;
#pragma unroll
        for (int t = 0; t < 16; ++t) gacc[t] += fast_tanh(bacc[t]) * gw[t];
    }
    // received (NG==1 mean) -> xin cols 64..127 ; cc-add only touches nodes < C (not output)
#pragma unroll
    for (int t = 0; t < 16; ++t) A[b * 132 + 64 + dq * 16 + t] = fast_tanh(gacc[t]);
    __syncthreads();

    wave_gemm64<128, true >(A, 132, iw1 + (size_t)n * 128 * 64, ib1 + (size_t)n * 64, Bf, 68, lane);
    __syncthreads();
    wave_gemm64<64, false>(Bf, 68, iw2 + (size_t)n * 64 * 64, ib2 + (size_t)n * 64, A, 132, lane);
    __syncthreads();
    // gated residual: h_new = g*mlp_out + (1-g)*h
    for (int i = lane; i < 512; i += 32) {
        const int bb = i >> 6, d = i & 63;
        const float g  = fast_sigmoid(igate[(size_t)n * 64 + d]);
        const float hn = g * A[bb * 132 + d] + (1.0f - g) * h[((size_t)bb * NN + n) * DD + d];
        A[bb * 132 + d] = hn;
    }
    __syncthreads();
    wave_gemm64<64, true>(A, 132, mw1 + (size_t)n * 64 * 64, mb1 + (size_t)n * 64, Bf, 68, lane);
    __syncthreads();
    wave_gemm64<64, true>(Bf, 68, mw2 + (size_t)n * 64 * 64, mb2 + (size_t)n * 64, A, 132, lane);
    __syncthreads();
    for (int i = lane; i < 512; i += 32) {
        const int bb = i >> 6, d = i & 63;
        out[((size_t)bb * 64 + node_off) * 64 + d] = A[bb * 132 + d];
    }
}

extern "C" void kernel_launch(void* const* d_in, const int* in_sizes, int n_in,
                              void* d_out, int out_size, void* d_ws, size_t ws_size,
                              hipStream_t stream) {
    (void)in_sizes; (void)n_in; (void)out_size; (void)ws_size;
    const float* h     = (const float*)d_in[1];
    const float* tp    = (const float*)d_in[2];
    const float* tk    = (const float*)d_in[3];
    const int*   conn  = (const int*)d_in[4];
    const unsigned char* cmask = (const unsigned char*)d_in[5];
    const float* key_p = (const float*)d_in[6];
    const float* dbw   = (const float*)d_in[7];
    const float* dgw   = (const float*)d_in[8];
    const float* iw1   = (const float*)d_in[9];
    const float* ib1   = (const float*)d_in[10];
    const float* iw2   = (const float*)d_in[11];
    const float* ib2   = (const float*)d_in[12];
    const float* igate = (const float*)d_in[13];
    const float* mw1   = (const float*)d_in[14];
    const float* mb1   = (const float*)d_in[15];
    const float* mw2   = (const float*)d_in[16];
    const float* mb2   = (const float*)d_in[17];
    const float* modw1 = (const float*)d_in[18];
    const float* modb1 = (const float*)d_in[19];
    const float* modw2 = (const float*)d_in[20];
    const float* modb2 = (const float*)d_in[21];
    const float* lrl   = (const float*)d_in[22];

    float* pm  = (float*)d_ws;                       // prev_msg: 8*4096*64 floats (8 MB)
    float* ekw = pm + (size_t)BB * NN * DD;          // eff_key:  64*8*64 floats (128 KB)

    k_prev_msg<<<NN / 4, 128, 0, stream>>>(h, mw1, mb1, mw2, mb2, pm);
    k_effkey<<<CC / 2, 64, 0, stream>>>(h, tp, tk, modw1, modb1, modw2, modb2, lrl, key_p, ekw);
    k_update<<<CC, 32, 0, stream>>>(h, conn, cmask, dbw, dgw, iw1, ib1, iw2, ib2, igate,
                                    mw1, mb1, mw2, mb2, pm, ekw, (float*)d_out);
}